// Net_84344567759246
// MI455X (gfx1250) — compile-verified
//
#include <hip/hip_runtime.h>
#include <math.h>

// f32 WMMA fragment types (wave32): A/B = 2 VGPRs, C/D = 8 VGPRs
typedef __attribute__((ext_vector_type(2))) float v2f;
typedef __attribute__((ext_vector_type(8))) float v8f;

#define NEG_SLOPE 0.01f

// ---------------- graph normalization ----------------
__global__ void deg_kernel(const int* __restrict__ row, const float* __restrict__ w,
                           float* __restrict__ deg, int E) {
  int e = blockIdx.x * blockDim.x + threadIdx.x;
  if (e < E) atomicAdd(&deg[row[e]], w[e]);
}

__global__ void dinv_kernel(const float* __restrict__ deg, float* __restrict__ dinv, int n) {
  int i = blockIdx.x * blockDim.x + threadIdx.x;
  if (i < n) {
    float d = deg[i];
    dinv[i] = (d > 0.0f) ? rsqrtf(d) : 0.0f;
  }
}

__global__ void norm_kernel(const int* __restrict__ row, const int* __restrict__ col,
                            const float* __restrict__ w, const float* __restrict__ dinv,
                            float* __restrict__ norm, int E) {
  int e = blockIdx.x * blockDim.x + threadIdx.x;
  if (e < E) norm[e] = -dinv[row[e]] * w[e] * dinv[col[e]];
}

// ---------------- SpMM: out[row[e],f] += norm[e]*x[col[e],f] ----------------
// one thread per (edge, feature). Lanes of a wave32 share one edge (F = 32/64),
// so row/col/norm loads are wave-uniform and the gather/scatter is a coalesced
// 128B row access; scatter resolves as global_atomic_add_f32 in L2 (feature
// tables are 12.8-25.6 MB -> L2 resident on the 192 MB L2).
__global__ void spmm_kernel(const int* __restrict__ row, const int* __restrict__ col,
                            const float* __restrict__ norm, const float* __restrict__ x,
                            float* __restrict__ out, long long total, int logF) {
  long long gid = (long long)blockIdx.x * blockDim.x + threadIdx.x;
  if (gid >= total) return;
  int e = (int)(gid >> logF);
  int f = (int)(gid & ((1 << logF) - 1));
  int F = 1 << logF;
  float v = norm[e] * x[(long long)col[e] * F + f];
  atomicAdd(&out[(long long)row[e] * F + f], v);
}

// ---------------- WMMA ChebConv GEMM ----------------
// out[n x FOUT] = A1[n x K] @ W1[K x FOUT] (+ A2 @ W2 if DUAL) + bias,
// optional leaky-relu. One wave32 per 16x16 output tile; K stepped by 4 with
// V_WMMA_F32_16X16X4_F32, fully unrolled.
// A layout (16x4 f32): lane&15 = M row; VGPR0/1 hold K + (lane>=16?2:0) + {0,1}
// C layout (16x16 f32): VGPR v -> M = v + (lane>=16?8:0), N = lane&15
// Column guard (only FOUT=2 needs it): clamp the load index instead of
// predicating — out-of-range lanes compute garbage C columns that are never
// stored, so EXEC stays all-ones for every WMMA and no saveexec is emitted.
// Row guard: single uniform full-tile test; full tiles emit an unguarded
// clause of 8 global_store_b32 with immediate offsets.
template <int K, int FOUT, bool RELU, bool DUAL>
__global__ void cheb_gemm_wmma(const float* __restrict__ A1, const float* __restrict__ W1,
                               const float* __restrict__ A2, const float* __restrict__ W2,
                               const float* __restrict__ bias, float* __restrict__ out,
                               int n) {
  const int lane = threadIdx.x;        // 0..31
  const int m15  = lane & 15;
  const int hi   = lane >> 4;          // 0 or 1
  const int row0 = blockIdx.x * 16;
  const int col0 = blockIdx.y * 16;
  const int gn   = col0 + m15;                                  // output column
  const int gnc  = (FOUT % 16 == 0) ? gn
                                    : (gn < FOUT ? gn : FOUT - 1);  // clamped

  v8f c = {};

#pragma unroll
  for (int k = 0; k < K; k += 4) {
    v2f a, b;
    const float* ar = A1 + (long long)(row0 + m15) * K + (k + hi * 2);
    a.x = ar[0];
    a.y = ar[1];
    const int kk = k + hi * 2;
    b.x = W1[(long long)kk * FOUT + gnc];
    b.y = W1[(long long)(kk + 1) * FOUT + gnc];
    c = __builtin_amdgcn_wmma_f32_16x16x4_f32(false, a, false, b, (short)0, c, false, false);
  }

  if (DUAL) {
#pragma unroll
    for (int k = 0; k < K; k += 4) {
      v2f a, b;
      const float* ar = A2 + (long long)(row0 + m15) * K + (k + hi * 2);
      a.x = ar[0];
      a.y = ar[1];
      const int kk = k + hi * 2;
      b.x = W2[(long long)kk * FOUT + gnc];
      b.y = W2[(long long)(kk + 1) * FOUT + gnc];
      c = __builtin_amdgcn_wmma_f32_16x16x4_f32(false, a, false, b, (short)0, c, false, false);
    }
  }

  // --- bias + activation (shared by both store paths) ---
  const float bv = bias[gnc];
  float r[8];
#pragma unroll
  for (int i = 0; i < 8; ++i) {
    float v = c[i] + bv;
    if (RELU) v = (v > 0.0f) ? v : NEG_SLOPE * v;
    r[i] = v;
  }

  const bool colOK = (FOUT % 16 == 0) || (gn < FOUT);
  float* outp = out + (long long)(row0 + hi * 8) * FOUT + gn;

  if (row0 + 16 <= n) {            // uniform: full tile -> store clause
    if (colOK) {
#pragma unroll
      for (int i = 0; i < 8; ++i) outp[(long long)i * FOUT] = r[i];
    }
  } else {                         // rare partial tail tile
    if (colOK) {
#pragma unroll
      for (int i = 0; i < 8; ++i)
        if (row0 + hi * 8 + i < n) outp[(long long)i * FOUT] = r[i];
    }
  }
}

// ---------------- mean pool (2 cols) ----------------
__global__ void reduce_kernel(const float* __restrict__ o3, float* __restrict__ sums, int n) {
  __shared__ float s0[256];
  __shared__ float s1[256];
  int i = blockIdx.x * blockDim.x + threadIdx.x;
  float a = 0.0f, b = 0.0f;
  if (i < n) { a = o3[2 * i]; b = o3[2 * i + 1]; }
  s0[threadIdx.x] = a;
  s1[threadIdx.x] = b;
  __syncthreads();
  for (int s = 128; s > 0; s >>= 1) {
    if (threadIdx.x < s) {
      s0[threadIdx.x] += s0[threadIdx.x + s];
      s1[threadIdx.x] += s1[threadIdx.x + s];
    }
    __syncthreads();
  }
  if (threadIdx.x == 0) {
    atomicAdd(&sums[0], s0[0]);
    atomicAdd(&sums[1], s1[0]);
  }
}

__global__ void finalize_kernel(const float* __restrict__ sums, float* __restrict__ out, float invn) {
  if (threadIdx.x == 0 && blockIdx.x == 0) {
    float a = sums[0] * invn;
    float b = sums[1] * invn;
    float m = fmaxf(a, b);
    float lse = m + logf(expf(a - m) + expf(b - m));
    out[0] = a - lse;
    out[1] = b - lse;
  }
}

extern "C" void kernel_launch(void* const* d_in, const int* in_sizes, int n_in,
                              void* d_out, int out_size, void* d_ws, size_t ws_size,
                              hipStream_t stream) {
  (void)n_in; (void)out_size; (void)ws_size;

  const float* x     = (const float*)d_in[0];
  const int*   ei    = (const int*)d_in[1];
  const float* attr  = (const float*)d_in[2];
  const float* w1_0  = (const float*)d_in[3];
  const float* b1    = (const float*)d_in[4];
  const float* w2_0  = (const float*)d_in[5];
  const float* w2_1  = (const float*)d_in[6];
  const float* b2    = (const float*)d_in[7];
  const float* w3_0  = (const float*)d_in[8];
  const float* w3_1  = (const float*)d_in[9];
  const float* b3    = (const float*)d_in[10];

  const int E = in_sizes[2];
  const int N = in_sizes[0] / 20;
  const int* row = ei;       // edge_index[0]
  const int* col = ei + E;   // edge_index[1]

  // workspace layout (floats)
  float* ws   = (float*)d_ws;
  float* deg  = ws;                 size_t off = (size_t)N;
  float* dinv = ws + off;           off += (size_t)N;
  float* norm = ws + off;           off += (size_t)E;
  float* h1   = ws + off;           off += (size_t)N * 32;   // layer1 out
  float* t2   = ws + off;           off += (size_t)N * 32;   // propagate(h1)
  float* h2   = ws + off;           off += (size_t)N * 64;   // layer2 out
  float* t3   = ws + off;           off += (size_t)N * 64;   // propagate(h2)
  float* o3   = ws + off;           off += (size_t)N * 2;    // layer3 out
  float* sums = ws + off;           off += 2;

  const int BT = 256;
  const int ebl = (E + BT - 1) / BT;
  const int nbl = (N + BT - 1) / BT;
  const int rtiles = (N + 15) / 16;

  // ---- normalization ----
  hipMemsetAsync(deg, 0, (size_t)N * sizeof(float), stream);
  deg_kernel<<<ebl, BT, 0, stream>>>(row, attr, deg, E);
  dinv_kernel<<<nbl, BT, 0, stream>>>(deg, dinv, N);
  norm_kernel<<<ebl, BT, 0, stream>>>(row, col, attr, dinv, norm, E);

  // ---- layer 1: h1 = leaky(x @ w1_0 + b1), K=20, FOUT=32 ----
  cheb_gemm_wmma<20, 32, true, false>
      <<<dim3(rtiles, 2), 32, 0, stream>>>(x, w1_0, nullptr, nullptr, b1, h1, N);

  // ---- layer 2: t2 = propagate(h1); h2 = leaky(h1@w2_0 + t2@w2_1 + b2) ----
  hipMemsetAsync(t2, 0, (size_t)N * 32 * sizeof(float), stream);
  {
    long long total = (long long)E * 32;
    long long blocks = (total + BT - 1) / BT;
    spmm_kernel<<<(unsigned)blocks, BT, 0, stream>>>(row, col, norm, h1, t2, total, 5);
  }
  cheb_gemm_wmma<32, 64, true, true>
      <<<dim3(rtiles, 4), 32, 0, stream>>>(h1, w2_0, t2, w2_1, b2, h2, N);

  // ---- layer 3: t3 = propagate(h2); o3 = h2@w3_0 + t3@w3_1 + b3 ----
  hipMemsetAsync(t3, 0, (size_t)N * 64 * sizeof(float), stream);
  {
    long long total = (long long)E * 64;
    long long blocks = (total + BT - 1) / BT;
    spmm_kernel<<<(unsigned)blocks, BT, 0, stream>>>(row, col, norm, h2, t3, total, 6);
  }
  cheb_gemm_wmma<64, 2, false, true>
      <<<dim3(rtiles, 1), 32, 0, stream>>>(h2, w3_0, t3, w3_1, b3, o3, N);

  // ---- mean pool + log_softmax ----
  hipMemsetAsync(sums, 0, 2 * sizeof(float), stream);
  reduce_kernel<<<nbl, BT, 0, stream>>>(o3, sums, N);
  finalize_kernel<<<1, 32, 0, stream>>>(sums, (float*)d_out, 1.0f / (float)N);
}